// ThreeOffsetsBlock_87660282511815
// MI455X (gfx1250) — compile-verified
//
#include <hip/hip_runtime.h>
#include <hip/hip_bf16.h>
#include <math.h>

typedef float v2f __attribute__((ext_vector_type(2)));
typedef float v8f __attribute__((ext_vector_type(8)));

#define NPOS (1u << 19)      // B*D*H*W = 2*64*64*64
#define DHW  (1u << 18)      // 64*64*64

// ---------------------------------------------------------------------------
// 1D trilinear blend factor (constant-offset deformable conv == dense conv
// with blended weights): tap -1 displaced +a -> (1-a)@-1, a@0 ; tap 0 -> 1@0 ;
// tap +1 displaced -a -> a@0, (1-a)@+1.
// ---------------------------------------------------------------------------
__device__ inline float Tlin(int t, int s, float a) {
  if (t == 0) return s == 0 ? 1.f - a : (s == 1 ? a : 0.f);
  if (t == 1) return s == 1 ? 1.f : 0.f;
  return s == 1 ? a : (s == 2 ? 1.f - a : 0.f);
}

// Pack concatenated conv weights as Wp[tap][co(192)][ci] (ci contiguous for
// float2 A-loads); effective deformable weights computed on the fly.
// Also packs concatenated bias / gamma / beta (192 each).
__global__ void pack_weights(const float* __restrict__ cbw, const float* __restrict__ daw,
                             const float* __restrict__ dbw,
                             const float* __restrict__ cbb, const float* __restrict__ dab,
                             const float* __restrict__ dbb,
                             const float* __restrict__ cbg, const float* __restrict__ dag,
                             const float* __restrict__ dbg,
                             const float* __restrict__ cbe, const float* __restrict__ dae,
                             const float* __restrict__ dbe,
                             int CIN, float* __restrict__ Wp, float* __restrict__ bcat,
                             float* __restrict__ gcat, float* __restrict__ becat) {
  int idx = blockIdx.x * 256 + threadIdx.x;
  if (idx < 192) {
    bcat[idx]  = idx < 64 ? cbb[idx] : (idx < 128 ? dab[idx - 64] : dbb[idx - 128]);
    gcat[idx]  = idx < 64 ? cbg[idx] : (idx < 128 ? dag[idx - 64] : dbg[idx - 128]);
    becat[idx] = idx < 64 ? cbe[idx] : (idx < 128 ? dae[idx - 64] : dbe[idx - 128]);
  }
  int total = 27 * 192 * CIN;
  if (idx >= total) return;
  int ci = idx % CIN;
  int co = (idx / CIN) % 192;
  int t  = idx / (CIN * 192);
  int p = t / 9, q = (t / 3) % 3, r = t % 3;
  float v;
  if (co < 64) {
    v = cbw[(co * CIN + ci) * 27 + t];
  } else {
    float a = co < 128 ? 0.4f : 0.7f;
    const float* w = co < 128 ? daw : dbw;
    int o = co & 63;
    v = 0.f;
    for (int ti = 0; ti < 3; ++ti) {
      float wi = Tlin(ti, p, a); if (wi == 0.f) continue;
      for (int tj = 0; tj < 3; ++tj) {
        float wj = Tlin(tj, q, a); if (wj == 0.f) continue;
        for (int tk = 0; tk < 3; ++tk) {
          float wk = Tlin(tk, r, a); if (wk == 0.f) continue;
          v += wi * wj * wk * w[(o * CIN + ci) * 27 + ti * 9 + tj * 3 + tk];
        }
      }
    }
  }
  Wp[idx] = v;
}

// ---------------------------------------------------------------------------
// Implicit-GEMM 3x3x3 conv, Cout=192, via V_WMMA_F32_16X16X4_F32.
// Block: 384 threads = 12 waves; grid = B*D*H rows. Each wave owns one
// 16-channel tile x the FULL 64-wide W row (4 accumulator tiles), so each
// A-operand float2 is reused by 4 WMMAs. The x-slab (16 ci x 3x3x66, stride-18
// padded) is staged in LDS once per ci-chunk and shared by all 12 waves.
// x layout NCDHW (f32); y layout [192][NPOS] (channel-major, pos contiguous).
// ---------------------------------------------------------------------------
template <int CIN>
__global__ __launch_bounds__(384) void conv3x3_wmma(const float* __restrict__ x,
                                                    const float* __restrict__ Wp,
                                                    const float* __restrict__ bcat,
                                                    float* __restrict__ y) {
  __shared__ float slab[594 * 18];  // [ (td*3+th)*66 + wo ][ ci(16) + 2 pad ]
  int bid = blockIdx.x;             // b*4096 + d*64 + h
  int h = bid & 63, d = (bid >> 6) & 63, b = bid >> 12;
  int tid = threadIdx.x;
  int wid = tid >> 5, lane = tid & 31;
  int lane15 = lane & 15;
  int khalf  = (lane >> 4) * 2;  // K pair selector (0 or 2)
  int m_off  = (lane >> 4) * 8;  // C/D row-half selector
  int co0 = wid * 16;
  int p0  = (b << 18) + (d << 12) + (h << 6);

  v8f acc[4];
#pragma unroll
  for (int v = 0; v < 8; ++v) {
    float bz = bcat[co0 + v + m_off];
    acc[0][v] = bz; acc[1][v] = bz; acc[2][v] = bz; acc[3][v] = bz;
  }

  for (int ci0 = 0; ci0 < CIN; ci0 += 16) {
    __syncthreads();
    // Stage slab: 9 (td,th) lines x 16 ci x 66 w (coalesced along w).
    for (int e = tid; e < 594 * 16; e += 384) {
      int wo = e % 66;
      int line = e / 66;          // g*16 + ci
      int ci = line & 15, g = line >> 4;
      int td = g / 3, th = g - td * 3;
      int dd = d + td - 1, hh = h + th - 1, ww = wo - 1;
      float v = 0.f;
      if ((unsigned)dd < 64u && (unsigned)hh < 64u && (unsigned)ww < 64u)
        v = x[((size_t)(b * CIN + ci0 + ci) << 18) + (dd << 12) + (hh << 6) + ww];
      slab[(g * 66 + wo) * 18 + ci] = v;
    }
    __syncthreads();
    for (int t = 0; t < 27; ++t) {
      int td = t / 9, th = (t / 3) % 3, tw = t % 3;
      const float* arow = &Wp[((size_t)(t * 192 + co0 + lane15)) * CIN + ci0 + khalf];
      const float* brow = &slab[((td * 3 + th) * 66 + tw + lane15) * 18 + khalf];
#pragma unroll
      for (int k4 = 0; k4 < 4; ++k4) {
        v2f av = *(const v2f*)(arow + k4 * 4);
#pragma unroll
        for (int ns = 0; ns < 4; ++ns) {
          v2f bv = *(const v2f*)(brow + ns * 288 + k4 * 4);  // ns*16*18
          acc[ns] = __builtin_amdgcn_wmma_f32_16x16x4_f32(false, av, false, bv,
                                                          (short)0, acc[ns],
                                                          false, false);
        }
      }
    }
  }
#pragma unroll
  for (int ns = 0; ns < 4; ++ns)
#pragma unroll
    for (int v = 0; v < 8; ++v)
      y[(size_t)(co0 + v + m_off) * NPOS + p0 + ns * 16 + lane15] = acc[ns][v];
}

// ---------------------------------------------------------------------------
// Per-channel BN stats over NPOS elements (double accumulation, deterministic
// fixed-order reduction). Emits fused scale/shift: xn = x*scale + shift.
// ---------------------------------------------------------------------------
__global__ void bn_stats(const float* __restrict__ data, const float* __restrict__ g,
                         const float* __restrict__ be, float* __restrict__ scale,
                         float* __restrict__ shift) {
  const int c = blockIdx.x;
  const float* p = data + (size_t)c * NPOS;
  double s1 = 0.0, s2 = 0.0;
  for (int i = threadIdx.x; i < (int)NPOS; i += 256) {
    double v = (double)p[i];
    s1 += v;
    s2 += v * v;
  }
  __shared__ double r1[256], r2[256];
  r1[threadIdx.x] = s1;
  r2[threadIdx.x] = s2;
  __syncthreads();
  for (int s = 128; s > 0; s >>= 1) {
    if (threadIdx.x < s) {
      r1[threadIdx.x] += r1[threadIdx.x + s];
      r2[threadIdx.x] += r2[threadIdx.x + s];
    }
    __syncthreads();
  }
  if (threadIdx.x == 0) {
    double m   = r1[0] / (double)NPOS;
    double var = r2[0] / (double)NPOS - m * m;
    float rstd = (float)(1.0 / sqrt(var + 1e-5));
    scale[c] = g[c] * rstd;
    shift[c] = be[c] - (float)m * g[c] * rstd;
  }
}

// ---------------------------------------------------------------------------
// 1x1x1 conv (192 -> 64) as WMMA GEMM. The 64-position x 192-channel input
// panel is staged in LDS ONCE per block with BN+ReLU applied during staging
// (layout [wo][cg], stride 196: cg-pairs contiguous for ds_load_b64, banks
// spread). 8 waves = 4 co-tiles x 2 position-pairs; each wave holds 2
// accumulator tiles so each A float2 feeds 2 WMMAs. u layout [64][NPOS].
// ---------------------------------------------------------------------------
__global__ __launch_bounds__(256) void gemm1x1_wmma(const float* __restrict__ y,
                                                    const float* __restrict__ W1,
                                                    const float* __restrict__ b1,
                                                    const float* __restrict__ scY,
                                                    const float* __restrict__ shY,
                                                    float* __restrict__ u) {
  __shared__ float sc[192], sh[192];
  __shared__ float bn[64 * 196];
  if (threadIdx.x < 192) {
    sc[threadIdx.x] = scY[threadIdx.x];
    sh[threadIdx.x] = shY[threadIdx.x];
  }
  __syncthreads();
  int p0 = blockIdx.x * 64;
  for (int e = threadIdx.x; e < 192 * 64; e += 256) {
    int cg = e >> 6, wo = e & 63;
    float yv = y[(size_t)cg * NPOS + p0 + wo];
    bn[wo * 196 + cg] = fmaxf(fmaf(yv, sc[cg], sh[cg]), 0.f);
  }
  __syncthreads();

  int wid = threadIdx.x >> 5, lane = threadIdx.x & 31;
  int lane15 = lane & 15, khalf = (lane >> 4) * 2, m_off = (lane >> 4) * 8;
  int co0 = (wid & 3) * 16;
  int nsA = (wid >> 2) * 2;  // first of this wave's two 16-pos subtiles

  v8f acc[2];
#pragma unroll
  for (int v = 0; v < 8; ++v) {
    float bz = b1[co0 + v + m_off];
    acc[0][v] = bz; acc[1][v] = bz;
  }

  for (int k4 = 0; k4 < 48; ++k4) {
    int cg = k4 * 4 + khalf;
    v2f av = *(const v2f*)&W1[(co0 + lane15) * 192 + cg];
#pragma unroll
    for (int ns = 0; ns < 2; ++ns) {
      v2f bv = *(const v2f*)&bn[((nsA + ns) * 16 + lane15) * 196 + cg];
      acc[ns] = __builtin_amdgcn_wmma_f32_16x16x4_f32(false, av, false, bv,
                                                      (short)0, acc[ns],
                                                      false, false);
    }
  }
#pragma unroll
  for (int ns = 0; ns < 2; ++ns)
#pragma unroll
    for (int v = 0; v < 8; ++v)
      u[(size_t)(co0 + v + m_off) * NPOS + p0 + (nsA + ns) * 16 + lane15] = acc[ns][v];
}

// Final BN+ReLU, transposing [64][NPOS] -> NCDHW.
__global__ void bn_relu_out(const float* __restrict__ u, const float* __restrict__ scale,
                            const float* __restrict__ shift, float* __restrict__ out) {
  size_t idx = (size_t)blockIdx.x * 256 + threadIdx.x;  // idx = c*NPOS + p
  int c = (int)(idx >> 19);
  int p = (int)(idx & (NPOS - 1));
  int b = p >> 18;
  int s = p & (DHW - 1);
  float v = fmaf(u[idx], scale[c], shift[c]);
  out[((size_t)(b * 64 + c) << 18) + s] = fmaxf(v, 0.f);
}

// ---------------------------------------------------------------------------
struct WS {
  float *Wp, *bcat, *gcat, *becat, *scY, *shY, *scU, *shU, *y, *u, *xs;
};

template <int CIN>
static void run_stage(const float* x, const float* const* P, float* out, const WS& ws,
                      hipStream_t s) {
  // P: 0 cb_w 1 cb_b 2 cb_g 3 cb_be | 4..7 d?a | 8..11 d?b | 12..15 ck
  int total = 27 * 192 * CIN;
  pack_weights<<<(total + 255) / 256, 256, 0, s>>>(
      P[0], P[4], P[8], P[1], P[5], P[9], P[2], P[6], P[10], P[3], P[7], P[11],
      CIN, ws.Wp, ws.bcat, ws.gcat, ws.becat);
  conv3x3_wmma<CIN><<<8192, 384, 0, s>>>(x, ws.Wp, ws.bcat, ws.y);
  bn_stats<<<192, 256, 0, s>>>(ws.y, ws.gcat, ws.becat, ws.scY, ws.shY);
  gemm1x1_wmma<<<8192, 256, 0, s>>>(ws.y, P[12], P[13], ws.scY, ws.shY, ws.u);
  bn_stats<<<64, 256, 0, s>>>(ws.u, P[14], P[15], ws.scU, ws.shU);
  bn_relu_out<<<(64u * NPOS) / 256, 256, 0, s>>>(ws.u, ws.scU, ws.shU, out);
}

extern "C" void kernel_launch(void* const* d_in, const int* in_sizes, int n_in,
                              void* d_out, int out_size, void* d_ws, size_t ws_size,
                              hipStream_t stream) {
  (void)in_sizes; (void)n_in; (void)out_size; (void)ws_size;
  const float* x = (const float*)d_in[0];
  const float* P1[16];
  const float* P2[16];
  for (int i = 0; i < 16; ++i) {
    P1[i] = (const float*)d_in[1 + i];
    P2[i] = (const float*)d_in[17 + i];
  }

  char* base = (char*)d_ws;
  size_t off = 0;
  auto carve = [&](size_t bytes) -> float* {
    float* r = (float*)(base + off);
    off = (off + bytes + 255) & ~(size_t)255;
    return r;
  };
  WS ws;
  ws.Wp    = carve((size_t)27 * 192 * 64 * 4);
  ws.bcat  = carve(192 * 4);
  ws.gcat  = carve(192 * 4);
  ws.becat = carve(192 * 4);
  ws.scY   = carve(192 * 4);
  ws.shY   = carve(192 * 4);
  ws.scU   = carve(64 * 4);
  ws.shU   = carve(64 * 4);
  ws.y     = carve((size_t)192 * NPOS * 4);  // 402.7 MB
  ws.u     = carve((size_t)64 * NPOS * 4);   // 134.2 MB
  ws.xs    = carve((size_t)64 * NPOS * 4);   // 134.2 MB (stage-1 output, NCDHW)

  run_stage<32>(x, P1, ws.xs, ws, stream);
  run_stage<64>(ws.xs, P2, (float*)d_out, ws, stream);
}